// Time2Vec_20761871909647
// MI455X (gfx1250) — compile-verified
//
#include <hip/hip_runtime.h>

typedef __attribute__((ext_vector_type(2))) float v2f;
typedef __attribute__((ext_vector_type(4))) float v4f;
typedef __attribute__((ext_vector_type(8))) float v8f;

// Time2Vec on gfx1250:
//   out[tok, d*64+e] = f( x[tok,d]*w[d,e] + b[d,e] ),  f = identity if e==0 else sin
// Implemented as one V_WMMA_F32_16X16X4_F32 per (16-token x 16-col) tile:
//   A (16x4): column K=0 = x[tok, d], K=1..3 = 0        (outer product trick)
//   B (4x16): row    K=0 = w[d, e..e+15], K=1..3 = 0
//   C (16x16): bias[d, e..e+15] broadcast over all 8 C VGPRs (bias is M-invariant)
// Store-bandwidth-bound: 128 MB fp32 out -> ~5.6 us floor at 23.3 TB/s.
__global__ __launch_bounds__(256) void time2vec_wmma_kernel(
    const float* __restrict__ x,    // [nTokens, 8]
    const float* __restrict__ w,    // [8, 64]
    const float* __restrict__ b,    // [8, 64]
    float* __restrict__ out,        // [nTokens, 512]
    int nTokens)
{
    const int lane    = (int)(threadIdx.x & 31u);
    const int waveId  = (int)(threadIdx.x >> 5u);
    const int group   = (int)blockIdx.x * 8 + waveId;   // one 16-token group per wave
    const int tokBase = group * 16;
    if (tokBase >= nTokens) return;                     // whole-wave uniform exit

    const int  lane16  = lane & 15;
    const bool lowHalf = lane < 16;

    // Each lane (mod 16) caches one token's 8 input features: two coalesced b128 loads.
    const v4f* xrow4 = (const v4f*)(x + (size_t)(tokBase + lane16) * 8);
    v4f x0 = xrow4[0];
    v4f x1 = xrow4[1];
    float xr[8] = { x0.x, x0.y, x0.z, x0.w, x1.x, x1.y, x1.z, x1.w };

    // D-matrix layout: VGPR v holds token M = v (lanes 0-15) / v+8 (lanes 16-31),
    // column N = lane%16.
    const int tokM0 = tokBase + (lane >> 4) * 8;
    float* outLane = out + (size_t)tokM0 * 512;

#pragma unroll
    for (int d = 0; d < 8; ++d) {
        // 16x4 f32 A layout: VGPR0 = K0 (lanes 0-15) / K2 (lanes 16-31), VGPR1 = K1/K3.
        v2f a;
        a.x = lowHalf ? xr[d] : 0.0f;   // only K=0 populated
        a.y = 0.0f;

#pragma unroll
        for (int t = 0; t < 4; ++t) {
            const int e  = t * 16 + lane16;        // column within this feature's block
            float wv = w[d * 64 + e];              // uniform per wave-half, L1-resident
            float bv = b[d * 64 + e];

            // 4x16 f32 B layout mirrors C/D striping: VGPR0 = K0 row (lanes 0-15).
            v2f bm;
            bm.x = lowHalf ? wv : 0.0f;            // only K=0 populated
            bm.y = 0.0f;

            v8f c = { bv, bv, bv, bv, bv, bv, bv, bv };   // bias, M-invariant

            v8f r = __builtin_amdgcn_wmma_f32_16x16x4_f32(
                /*neg_a=*/false, a, /*neg_b=*/false, bm,
                /*c_mod=*/(short)0, c, /*reuse_a=*/false, /*reuse_b=*/false);

            const bool linear = (t == 0) && (lane16 == 0);  // e == 0 stays affine
            float* p = outLane + d * 64 + e;
#pragma unroll
            for (int v = 0; v < 8; ++v) {
                float rv = r[v];
                // per-lane select (v_cndmask), no EXEC divergence
                p[(size_t)v * 512] = linear ? rv : __sinf(rv);
            }
        }
    }
}

extern "C" void kernel_launch(void* const* d_in, const int* in_sizes, int n_in,
                              void* d_out, int out_size, void* d_ws, size_t ws_size,
                              hipStream_t stream) {
    const float* x = (const float*)d_in[0];   // [32, 2048, 8]
    const float* w = (const float*)d_in[1];   // [8, 64]
    const float* b = (const float*)d_in[2];   // [8, 64]
    float* out = (float*)d_out;               // [32, 2048, 512]

    const int nTokens = in_sizes[0] / 8;      // 65536
    const int groups  = (nTokens + 15) / 16;  // 16 tokens per wave
    const int blocks  = (groups + 7) / 8;     // 8 waves per 256-thread block

    time2vec_wmma_kernel<<<blocks, 256, 0, stream>>>(x, w, b, out, nTokens);
}